// SELayer_36996848287772
// MI455X (gfx1250) — compile-verified
//
#include <hip/hip_runtime.h>
#include <hip/hip_bf16.h>

typedef __attribute__((ext_vector_type(2))) float v2f;
typedef __attribute__((ext_vector_type(8))) float v8f;

constexpr int NB  = 32;    // batch
constexpr int NC  = 256;   // channels
constexpr int NH  = 16;    // hidden
constexpr int HW  = 4096;  // 64*64 spatial
constexpr int BD  = 256;   // hidden^2 (bilinear dim)

// ---------------------------------------------------------------------------
// Kernel 1: global average pool.  One 256-thread block per (b,c) pair.
// Each thread does 4 coalesced float4 loads (16 floats), LDS tree-reduce.
// Default RT temporal hint so x stays resident in the 192MB L2 for kernel 3.
// ---------------------------------------------------------------------------
__global__ void se_pool_kernel(const float* __restrict__ x,
                               float* __restrict__ pooled) {
    __shared__ float red[256];
    const int bc = blockIdx.x;                       // 0 .. 8191
    const float4* xv = (const float4*)(x + (size_t)bc * HW);
    const int t = threadIdx.x;
    float s = 0.0f;
#pragma unroll
    for (int i = 0; i < 4; ++i) {
        float4 v = xv[t + 256 * i];
        s += (v.x + v.y) + (v.z + v.w);
    }
    red[t] = s;
    __syncthreads();
#pragma unroll
    for (int off = 128; off > 0; off >>= 1) {
        if (t < off) red[t] += red[t + off];
        __syncthreads();
    }
    if (t == 0) pooled[bc] = red[0] * (1.0f / 4096.0f);
}

// ---------------------------------------------------------------------------
// Kernel 2: squeeze path, single 1024-thread block (32 wave32 waves).
//   Phase 1: y = relu(pooled @ w1^T)            [32 x 16]   (512 dot products)
//   Phase 2: outer[b][i*16+j] = y[b][i]*y[b][j] [32 x 256]  -> LDS (32 KB)
//   Phase 3: s = sigmoid(outer @ w2^T)          [32 x 256]  via WMMA f32 16x16x4
// WMMA tiling: M=32 (2 tiles) x N=256 (16 tiles) = 32 tiles, one wave each,
// K=256 in 64 steps of 4.  EXEC is all-ones at every WMMA (uniform control).
// ---------------------------------------------------------------------------
__global__ void se_squeeze_kernel(const float* __restrict__ pooled,
                                  const float* __restrict__ w1,
                                  const float* __restrict__ w2,
                                  float* __restrict__ s_out) {
    __shared__ float y[NB * NH];        // 2 KB
    __shared__ float outer[NB * BD];    // 32 KB
    const int t = threadIdx.x;          // 0..1023

    // Phase 1: fc1 + relu
    if (t < NB * NH) {
        const int b = t >> 4, h = t & 15;
        const float* pr = pooled + b * NC;
        const float* wr = w1 + h * NC;
        float acc = 0.0f;
#pragma unroll 8
        for (int c = 0; c < NC; ++c) acc = fmaf(pr[c], wr[c], acc);
        y[t] = acc > 0.0f ? acc : 0.0f;
    }
    __syncthreads();

    // Phase 2: batched outer product into LDS
#pragma unroll
    for (int i = 0; i < 8; ++i) {
        const int idx = t + 1024 * i;            // 0..8191
        const int b = idx >> 8;
        const int d = idx & 255;
        outer[idx] = y[b * NH + (d >> 4)] * y[b * NH + (d & 15)];
    }
    __syncthreads();

    // Phase 3: fc2 via V_WMMA_F32_16X16X4_F32
    const int wave  = t >> 5;          // 0..31
    const int lane  = t & 31;
    const int mt    = wave >> 4;       // 0..1   (M tile)
    const int nt    = wave & 15;       // 0..15  (N tile)
    const int l16   = lane & 15;       // M row (for A) / N col (for B, D)
    const int khalf = lane >> 4;       // 0: K={k,k+1}, 1: K={k+2,k+3}

    // A[M][K] = outer[mt*16 + M][K]          (LDS)
    // B[K][N] = w2[nt*16 + N][K]             (global, row-major [256][256])
    const float* aptr = outer + (mt * 16 + l16) * BD;
    const float* bptr = w2 + (size_t)(nt * 16 + l16) * BD;

    v8f acc = {};
    for (int k = 0; k < BD; k += 4) {
        const int kk = k + 2 * khalf;
        v2f a; a.x = aptr[kk]; a.y = aptr[kk + 1];
        v2f bb; bb.x = bptr[kk]; bb.y = bptr[kk + 1];
        acc = __builtin_amdgcn_wmma_f32_16x16x4_f32(
            /*neg_a=*/false, a, /*neg_b=*/false, bb,
            /*c_mod=*/(short)0, acc, /*reuse_a=*/false, /*reuse_b=*/false);
    }

    // D layout: VGPR v -> M = v + 8*khalf, N = l16.  Apply sigmoid, store.
#pragma unroll
    for (int v = 0; v < 8; ++v) {
        const int bi = mt * 16 + khalf * 8 + v;
        const int ci = nt * 16 + l16;
        const float val = acc[v];
        s_out[bi * NC + ci] = 1.0f / (1.0f + __expf(-val));
    }
}

// ---------------------------------------------------------------------------
// Kernel 3: channel-wise rescale.  One 256-thread block per (b,c); x re-read
// should hit L2 (134 MB working set < 192 MB L2).  float4 in, float4 out.
// ---------------------------------------------------------------------------
__global__ void se_scale_kernel(const float* __restrict__ x,
                                const float* __restrict__ s,
                                float* __restrict__ out) {
    const int bc = blockIdx.x;
    const float sv = s[bc];
    const float4* xv = (const float4*)(x + (size_t)bc * HW);
    float4* ov = (float4*)(out + (size_t)bc * HW);
    const int t = threadIdx.x;
#pragma unroll
    for (int i = 0; i < 4; ++i) {
        float4 v = xv[t + 256 * i];
        v.x *= sv; v.y *= sv; v.z *= sv; v.w *= sv;
        ov[t + 256 * i] = v;
    }
}

extern "C" void kernel_launch(void* const* d_in, const int* in_sizes, int n_in,
                              void* d_out, int out_size, void* d_ws, size_t ws_size,
                              hipStream_t stream) {
    (void)in_sizes; (void)n_in; (void)out_size; (void)ws_size;
    const float* x  = (const float*)d_in[0];   // [32,256,64,64]
    const float* w1 = (const float*)d_in[1];   // [16,256]
    const float* w2 = (const float*)d_in[2];   // [256,256]
    float* out = (float*)d_out;

    float* pooled = (float*)d_ws;              // 8192 floats
    float* s      = pooled + NB * NC;          // 8192 floats

    se_pool_kernel<<<NB * NC, 256, 0, stream>>>(x, pooled);
    se_squeeze_kernel<<<1, 1024, 0, stream>>>(pooled, w1, w2, s);
    se_scale_kernel<<<NB * NC, 256, 0, stream>>>(x, s, out);
}